// Update_Node_80711025426903
// MI455X (gfx1250) — compile-verified
//
#include <hip/hip_runtime.h>
#include <hip/hip_bf16.h>
#include <stdint.h>

// ---------------- problem constants ----------------
#define DD   64          // MSA depth
#define LL   512         // sequence length
#define CC   256         // MSA channels
#define PCC  256         // pair channels
#define GG   4           // groups
#define GPP  64          // group planes
#define FFH  1024        // FFN hidden
#define ROWS (DD * LL)   // 32768
#define EPSL 1e-5f

typedef float v2f __attribute__((ext_vector_type(2)));
typedef float v8f __attribute__((ext_vector_type(8)));

static __device__ __forceinline__ v2f mk2(float a, float b) {
    v2f r; r.x = a; r.y = b; return r;
}

// ---------------------------------------------------------------------------
// LayerNorm over last dim (256), one wave (32 lanes) per row, 8 rows / block.
// ---------------------------------------------------------------------------
__global__ __launch_bounds__(256)
void ln_rows_kernel(const float* __restrict__ x, const float* __restrict__ w,
                    const float* __restrict__ b, float* __restrict__ y)
{
    int tid  = threadIdx.x;
    int lane = tid & 31;
    int wid  = tid >> 5;
    long row = (long)blockIdx.x * 8 + wid;
    int c0   = lane * 8;

    const float4* px = (const float4*)(x + row * CC + c0);
    float4 v0 = px[0];
    float4 v1 = px[1];

    float s  = v0.x + v0.y + v0.z + v0.w + v1.x + v1.y + v1.z + v1.w;
    float ss = v0.x*v0.x + v0.y*v0.y + v0.z*v0.z + v0.w*v0.w
             + v1.x*v1.x + v1.y*v1.y + v1.z*v1.z + v1.w*v1.w;
    #pragma unroll
    for (int off = 16; off > 0; off >>= 1) {
        s  += __shfl_xor(s,  off);
        ss += __shfl_xor(ss, off);
    }
    float mu  = s  * (1.0f / CC);
    float var = ss * (1.0f / CC) - mu * mu;
    float rs  = rsqrtf(var + EPSL);

    float4 w0 = ((const float4*)(w + c0))[0];
    float4 w1 = ((const float4*)(w + c0))[1];
    float4 b0 = ((const float4*)(b + c0))[0];
    float4 b1 = ((const float4*)(b + c0))[1];

    float4 o0, o1;
    o0.x = (v0.x - mu) * rs * w0.x + b0.x;
    o0.y = (v0.y - mu) * rs * w0.y + b0.y;
    o0.z = (v0.z - mu) * rs * w0.z + b0.z;
    o0.w = (v0.w - mu) * rs * w0.w + b0.w;
    o1.x = (v1.x - mu) * rs * w1.x + b1.x;
    o1.y = (v1.y - mu) * rs * w1.y + b1.y;
    o1.z = (v1.z - mu) * rs * w1.z + b1.z;
    o1.w = (v1.w - mu) * rs * w1.w + b1.w;

    float4* py = (float4*)(y + row * CC + c0);
    py[0] = o0;
    py[1] = o1;
}

// ---------------------------------------------------------------------------
// Attention logits: for each pair i<=j, symmetrize + layernorm + dot with WA.
// logits are symmetric in (i,j): write both.  One wave per pair.
// ---------------------------------------------------------------------------
__global__ __launch_bounds__(256)
void logits_kernel(const float* __restrict__ rr, const float* __restrict__ lnw,
                   const float* __restrict__ lnb, const float* __restrict__ WA,
                   const float* __restrict__ bA, float* __restrict__ logits)
{
    const long P = (long)LL * (LL + 1) / 2;   // 131328
    int tid  = threadIdx.x;
    int lane = tid & 31;
    int wid  = tid >> 5;
    long p   = (long)blockIdx.x * 8 + wid;
    if (p >= P) return;

    // decode upper-triangular (row-major incl diagonal) index p -> (i, j)
    double disc = (double)(2 * LL + 1) * (2 * LL + 1) - 8.0 * (double)p;
    int i = (int)(((2 * LL + 1) - sqrt(disc)) * 0.5);
    if (i < 0) i = 0;
    if (i > LL - 1) i = LL - 1;
    #define FTRI(ii) ((long)(ii) * (2 * LL + 1 - (ii)) / 2)
    while (i > 0 && FTRI(i) > p) --i;
    while (FTRI(i + 1) <= p) ++i;
    int j = i + (int)(p - FTRI(i));
    #undef FTRI

    int c0 = lane * 8;
    const float4* pa = (const float4*)(rr + ((long)i * LL + j) * PCC + c0);
    const float4* pb = (const float4*)(rr + ((long)j * LL + i) * PCC + c0);
    float4 a0 = pa[0], a1 = pa[1];
    float4 b0 = pb[0], b1 = pb[1];

    float e[8];
    e[0] = 0.5f * (a0.x + b0.x); e[1] = 0.5f * (a0.y + b0.y);
    e[2] = 0.5f * (a0.z + b0.z); e[3] = 0.5f * (a0.w + b0.w);
    e[4] = 0.5f * (a1.x + b1.x); e[5] = 0.5f * (a1.y + b1.y);
    e[6] = 0.5f * (a1.z + b1.z); e[7] = 0.5f * (a1.w + b1.w);

    float s = 0.f, ss = 0.f;
    #pragma unroll
    for (int k = 0; k < 8; ++k) { s += e[k]; ss += e[k] * e[k]; }
    #pragma unroll
    for (int off = 16; off > 0; off >>= 1) {
        s  += __shfl_xor(s,  off);
        ss += __shfl_xor(ss, off);
    }
    float mu  = s  * (1.0f / PCC);
    float var = ss * (1.0f / PCC) - mu * mu;
    float rs  = rsqrtf(var + EPSL);

    float4 w0 = ((const float4*)(lnw + c0))[0];
    float4 w1 = ((const float4*)(lnw + c0))[1];
    float4 lb0 = ((const float4*)(lnb + c0))[0];
    float4 lb1 = ((const float4*)(lnb + c0))[1];
    float wv[8] = {w0.x, w0.y, w0.z, w0.w, w1.x, w1.y, w1.z, w1.w};
    float bv[8] = {lb0.x, lb0.y, lb0.z, lb0.w, lb1.x, lb1.y, lb1.z, lb1.w};

    float lg[4] = {0.f, 0.f, 0.f, 0.f};
    #pragma unroll
    for (int k = 0; k < 8; ++k) {
        float nv = (e[k] - mu) * rs * wv[k] + bv[k];
        float4 wa = ((const float4*)WA)[c0 + k];   // WA row c -> 4 floats
        lg[0] += nv * wa.x; lg[1] += nv * wa.y;
        lg[2] += nv * wa.z; lg[3] += nv * wa.w;
    }
    #pragma unroll
    for (int off = 16; off > 0; off >>= 1) {
        lg[0] += __shfl_xor(lg[0], off);
        lg[1] += __shfl_xor(lg[1], off);
        lg[2] += __shfl_xor(lg[2], off);
        lg[3] += __shfl_xor(lg[3], off);
    }
    if (lane == 0) {
        float4 r;
        r.x = lg[0] + bA[0]; r.y = lg[1] + bA[1];
        r.z = lg[2] + bA[2]; r.w = lg[3] + bA[3];
        ((float4*)logits)[(long)i * LL + j] = r;
        ((float4*)logits)[(long)j * LL + i] = r;
    }
}

// ---------------------------------------------------------------------------
// Softmax over j for each (i,g). Output layout [G][L][L] (row-major in j),
// so each g-slice is directly the A-operand of the aggregation GEMM.
// ---------------------------------------------------------------------------
__global__ __launch_bounds__(256)
void softmax_kernel(const float* __restrict__ logits, float* __restrict__ Aout)
{
    __shared__ float4 sred[8];
    int i    = blockIdx.x;
    int tid  = threadIdx.x;
    int lane = tid & 31;
    int wid  = tid >> 5;

    float4 l1 = ((const float4*)logits)[(long)i * LL + tid];
    float4 l2 = ((const float4*)logits)[(long)i * LL + tid + 256];

    float4 mx;
    mx.x = fmaxf(l1.x, l2.x); mx.y = fmaxf(l1.y, l2.y);
    mx.z = fmaxf(l1.z, l2.z); mx.w = fmaxf(l1.w, l2.w);
    #pragma unroll
    for (int off = 16; off > 0; off >>= 1) {
        mx.x = fmaxf(mx.x, __shfl_xor(mx.x, off));
        mx.y = fmaxf(mx.y, __shfl_xor(mx.y, off));
        mx.z = fmaxf(mx.z, __shfl_xor(mx.z, off));
        mx.w = fmaxf(mx.w, __shfl_xor(mx.w, off));
    }
    if (lane == 0) sred[wid] = mx;
    __syncthreads();
    float4 M = sred[0];
    #pragma unroll
    for (int k = 1; k < 8; ++k) {
        float4 t = sred[k];
        M.x = fmaxf(M.x, t.x); M.y = fmaxf(M.y, t.y);
        M.z = fmaxf(M.z, t.z); M.w = fmaxf(M.w, t.w);
    }
    __syncthreads();

    float4 e1, e2;
    e1.x = expf(l1.x - M.x); e1.y = expf(l1.y - M.y);
    e1.z = expf(l1.z - M.z); e1.w = expf(l1.w - M.w);
    e2.x = expf(l2.x - M.x); e2.y = expf(l2.y - M.y);
    e2.z = expf(l2.z - M.z); e2.w = expf(l2.w - M.w);

    float4 s;
    s.x = e1.x + e2.x; s.y = e1.y + e2.y;
    s.z = e1.z + e2.z; s.w = e1.w + e2.w;
    #pragma unroll
    for (int off = 16; off > 0; off >>= 1) {
        s.x += __shfl_xor(s.x, off);
        s.y += __shfl_xor(s.y, off);
        s.z += __shfl_xor(s.z, off);
        s.w += __shfl_xor(s.w, off);
    }
    if (lane == 0) sred[wid] = s;
    __syncthreads();
    float4 S = sred[0];
    #pragma unroll
    for (int k = 1; k < 8; ++k) {
        float4 t = sred[k];
        S.x += t.x; S.y += t.y; S.z += t.z; S.w += t.w;
    }
    float4 inv;
    inv.x = 1.0f / S.x; inv.y = 1.0f / S.y;
    inv.z = 1.0f / S.z; inv.w = 1.0f / S.w;

    const long LL2 = (long)LL * LL;
    long base = (long)i * LL + tid;
    Aout[0 * LL2 + base]       = e1.x * inv.x;
    Aout[1 * LL2 + base]       = e1.y * inv.y;
    Aout[2 * LL2 + base]       = e1.z * inv.z;
    Aout[3 * LL2 + base]       = e1.w * inv.w;
    Aout[0 * LL2 + base + 256] = e2.x * inv.x;
    Aout[1 * LL2 + base + 256] = e2.y * inv.y;
    Aout[2 * LL2 + base + 256] = e2.z * inv.z;
    Aout[3 * LL2 + base + 256] = e2.w * inv.w;
}

// ---------------------------------------------------------------------------
// fp32 WMMA GEMM: C = A[M,K] * B[K,N] (+bias) (+resid) (relu), all row-major.
// Block: 256 threads = 8 waves -> 128(M) x 64(N) tile.
// B panel staged in LDS *k-paired*: Bs2[kp][n] = {B[2kp][n], B[2kp+1][n]}
// as native v2f, row stride 80 v2f (160 dwords == 32 mod 64 banks) so the
// lane<16 (kp) and lane>=16 (kp+1) halves hit disjoint bank groups, and each
// B fragment is ONE aligned ds_load_b64 straight into the WMMA operand pair
// (no v_mov packing).
// Each wave: one 16x64 strip, 4 x v_wmma_f32_16x16x4_f32 per K-step.
// EPI: 0 = none, 1 = +bias, 2 = +bias+resid, 3 = relu(+bias)
// 2-level batch: batch = blockIdx.y, b1 = batch/nb2, b2 = batch%nb2.
// ---------------------------------------------------------------------------
#define BSTRIDE 80   // v2f stride per kp row

template <int EPI>
__global__ __launch_bounds__(256)
void gemm_f32_wmma(const float* __restrict__ A, int lda, long sA1, long sA2,
                   const float* __restrict__ B, int ldb, long sB1, long sB2,
                   float* __restrict__ Cout, int ldc, long sC1, long sC2,
                   const float* __restrict__ bias,
                   const float* __restrict__ resid, long sR1, long sR2,
                   int K, int mblocks, int nb2)
{
    __shared__ v2f Bs2[32 * BSTRIDE];   // 32 kp rows x 64 cols (padded), 20 KB
    int tid  = threadIdx.x;
    int lane = tid & 31;
    int wid  = tid >> 5;
    int bx   = blockIdx.x;
    int mtb  = bx % mblocks;
    int nt   = bx / mblocks;
    int bat  = blockIdx.y;
    int b1   = bat / nb2;
    int b2   = bat % nb2;

    const float* Ab = A + (long)b1 * sA1 + (long)b2 * sA2;
    const float* Bb = B + (long)b1 * sB1 + (long)b2 * sB2;
    float*       Cb = Cout + (long)b1 * sC1 + (long)b2 * sC2;
    const float* Rb = resid ? (resid + (long)b1 * sR1 + (long)b2 * sR2) : nullptr;

    int  row0   = (mtb * 8 + wid) * 16;
    int  mlane  = lane & 15;
    int  half   = lane >> 4;                 // 0 / 1
    int  koff   = half << 1;                 // 0 / 2
    long aoff   = (long)(row0 + mlane) * lda;
    int  ncol0  = nt * 64;

    // staging indices: thread -> (kp, n0) ; kp in [0,32), n0 in {0,8,...,56}
    int s_kp = tid >> 3;
    int s_n0 = (tid & 7) * 8;

    v8f acc[4] = {};

    for (int kc = 0; kc < K; kc += 64) {
        // cooperative k-paired stage of B panel [64 x 64] into LDS
        {
            const float* g0 = Bb + (long)(kc + 2 * s_kp) * ldb + ncol0 + s_n0;
            const float* g1 = g0 + ldb;
            float4 u0 = *(const float4*)(g0);
            float4 u1 = *(const float4*)(g0 + 4);
            float4 t0 = *(const float4*)(g1);
            float4 t1 = *(const float4*)(g1 + 4);
            v2f* dst = &Bs2[s_kp * BSTRIDE + s_n0];
            dst[0] = mk2(u0.x, t0.x);
            dst[1] = mk2(u0.y, t0.y);
            dst[2] = mk2(u0.z, t0.z);
            dst[3] = mk2(u0.w, t0.w);
            dst[4] = mk2(u1.x, t1.x);
            dst[5] = mk2(u1.y, t1.y);
            dst[6] = mk2(u1.z, t1.z);
            dst[7] = mk2(u1.w, t1.w);
        }
        __syncthreads();

        #pragma unroll
        for (int k8 = 0; k8 < 64; k8 += 4) {
            float2 a2 = *(const float2*)(Ab + aoff + kc + k8 + koff);
            v2f av = mk2(a2.x, a2.y);

            const v2f* bp = &Bs2[((k8 >> 1) + half) * BSTRIDE + mlane];
            v2f bv0 = bp[0];
            v2f bv1 = bp[16];
            v2f bv2 = bp[32];
            v2f bv3 = bp[48];

            acc[0] = __builtin_amdgcn_wmma_f32_16x16x4_f32(false, av, false, bv0,
                                                           (short)0, acc[0], false, false);
            acc[1] = __builtin_amdgcn_wmma_f32_16x16x4_f32(false, av, false, bv1,
                                                           (short)0, acc[1], false, false);
            acc[2] = __builtin_amdgcn_wmma_f32_16x16x4_f32(false, av, false, bv2,
                                                           (short)0, acc[2], false, false);
            acc[3] = __builtin_amdgcn_wmma_f32_16x16x4_f32(false, av, false, bv3,
                                                           (short)0, acc[3], false, false);
        }
        __syncthreads();
    }

    // epilogue: VGPR r of a 16x16 tile holds rows (r, r+8) split by lane half
    int rbase = half * 8;
    #pragma unroll
    for (int j = 0; j < 4; ++j) {
        int   coln = ncol0 + j * 16 + mlane;
        float bb   = (EPI >= 1) ? bias[coln] : 0.0f;
        #pragma unroll
        for (int r = 0; r < 8; ++r) {
            long row = row0 + rbase + r;
            long off = row * (long)ldc + coln;
            float v  = acc[j][r] + bb;
            if constexpr (EPI == 2) v += Rb[off];
            if constexpr (EPI == 3) v = fmaxf(v, 0.0f);
            Cb[off] = v;
        }
    }
}

// ---------------------------------------------------------------------------
// launch
// ---------------------------------------------------------------------------
extern "C" void kernel_launch(void* const* d_in, const int* in_sizes, int n_in,
                              void* d_out, int out_size, void* d_ws, size_t ws_size,
                              hipStream_t stream)
{
    const float* msa    = (const float*)d_in[0];
    const float* rr     = (const float*)d_in[1];
    const float* lnm_w  = (const float*)d_in[2];
    const float* lnm_b  = (const float*)d_in[3];
    const float* lnr_w  = (const float*)d_in[4];
    const float* lnr_b  = (const float*)d_in[5];
    const float* WA     = (const float*)d_in[6];
    const float* bA     = (const float*)d_in[7];
    const float* WV     = (const float*)d_in[8];
    const float* bV     = (const float*)d_in[9];
    const float* WO     = (const float*)d_in[10];
    const float* bO     = (const float*)d_in[11];
    const float* lnf_w  = (const float*)d_in[12];
    const float* lnf_b  = (const float*)d_in[13];
    const float* W1     = (const float*)d_in[14];
    const float* b1     = (const float*)d_in[15];
    const float* W2     = (const float*)d_in[16];
    const float* b2     = (const float*)d_in[17];
    float* out = (float*)d_out;

    char* ws = (char*)d_ws;
    const size_t SZ_ACT = (size_t)ROWS * CC * sizeof(float);        // 32 MB
    float* m_buf   = (float*)(ws);                                  // m, later h
    float* v_buf   = (float*)(ws + SZ_ACT);                         // V
    float* agg_buf = (float*)(ws + 2 * SZ_ACT);                     // agg
    float* x_buf   = (float*)(ws + 3 * SZ_ACT);                     // x
    float* lg_buf  = (float*)(ws + 4 * SZ_ACT);                     // logits [L,L,G]
    float* a_buf   = (float*)(ws + 4 * SZ_ACT + (size_t)LL * LL * GG * 4); // A [G,L,L]
    float* t_buf   = (float*)(ws + 4 * SZ_ACT + 2 * (size_t)LL * LL * GG * 4); // FFN hidden

    const long LL2 = (long)LL * LL;
    const long LC  = (long)LL * CC;

    // 1) m = LN(msa)
    ln_rows_kernel<<<ROWS / 8, 256, 0, stream>>>(msa, lnm_w, lnm_b, m_buf);

    // 2) V = m @ WV + bV           [32768,256] x [256,256]
    gemm_f32_wmma<1><<<dim3(256 * 4, 1), 256, 0, stream>>>(
        m_buf, CC, 0, 0,  WV, CC, 0, 0,  v_buf, CC, 0, 0,
        bV, nullptr, 0, 0,  CC, 256, 1);

    // 3) symmetric attention logits (one pair-tensor read total)
    {
        long P = (long)LL * (LL + 1) / 2;
        int blocks = (int)((P + 7) / 8);
        logits_kernel<<<blocks, 256, 0, stream>>>(rr, lnr_w, lnr_b, WA, bA, lg_buf);
    }

    // 4) softmax over j -> A [G][L][L]
    softmax_kernel<<<LL, 256, 0, stream>>>(lg_buf, a_buf);

    // 5) agg[d,i,g*64+c] = sum_j A_g[i,j] V[d,j,g*64+c]   batched over (d,g)
    //    per batch: M=512, K=512, N=64
    gemm_f32_wmma<0><<<dim3(4, DD * GG), 256, 0, stream>>>(
        a_buf, LL, 0, LL2,          // A: g-slice stride L*L (b2 = g)
        v_buf, CC, LC, GPP,         // B: d stride L*C (b1 = d), g offset 64
        agg_buf, CC, LC, GPP,       // C: same batching as B
        nullptr, nullptr, 0, 0,  LL, 4, GG);

    // 6) x = msa + agg @ WO + bO
    gemm_f32_wmma<2><<<dim3(256 * 4, 1), 256, 0, stream>>>(
        agg_buf, CC, 0, 0,  WO, CC, 0, 0,  x_buf, CC, 0, 0,
        bO, msa, 0, 0,  CC, 256, 1);

    // 7) h = LN(x)   (reuses m_buf)
    ln_rows_kernel<<<ROWS / 8, 256, 0, stream>>>(x_buf, lnf_w, lnf_b, m_buf);

    // 8) t = relu(h @ W1 + b1)     [32768,256] x [256,1024]
    gemm_f32_wmma<3><<<dim3(256 * 16, 1), 256, 0, stream>>>(
        m_buf, CC, 0, 0,  W1, FFH, 0, 0,  t_buf, FFH, 0, 0,
        b1, nullptr, 0, 0,  CC, 256, 1);

    // 9) out = x + t @ W2 + b2     [32768,1024] x [1024,256]
    gemm_f32_wmma<2><<<dim3(256 * 4, 1), 256, 0, stream>>>(
        t_buf, FFH, 0, 0,  W2, CC, 0, 0,  out, CC, 0, 0,
        b2, x_buf, 0, 0,  FFH, 256, 1);
}